// PrototypeLibrary_17136919511078
// MI455X (gfx1250) — compile-verified
//
#include <hip/hip_runtime.h>
#include <hip/hip_bf16.h>

// Problem constants (match reference)
#define Bsz   4096
#define Ncl   64
#define Dd    128
#define EPSV  1e-3f
#define MINC  10.0f

#define SLICES 4          // B-dimension split for kernel A occupancy
#define CHUNKA 32         // rows of B staged per LDS chunk in kernel A (double-buffered)
#define EPITCH 136        // LDS pitch (floats) for embedding chunk = 128 + 8 pad DWORDs
#define IPITCH 136        // LDS pitch for inv chunk in kernel C
#define DPITCH 36         // LDS pitch for diff chunk in kernel C = 32 + 4 pad DWORDs
#define AUGP   257        // augmented-matrix pitch in kernel B

typedef float v2f  __attribute__((ext_vector_type(2)));
typedef float v8f  __attribute__((ext_vector_type(8)));
typedef unsigned int u32x4 __attribute__((ext_vector_type(4)));
typedef int   i32x4v __attribute__((ext_vector_type(4)));
typedef int   i32x8v __attribute__((ext_vector_type(8)));

// ---------------------------------------------------------------------------
// Tensor Data Mover: 2D tile (tile_w elems x tile_h rows, row stride in elems)
// from global -> LDS, with LDS padding so row pitch = tile_w + pad DWORDs.
//   pad_int_code: interval = 2^(code+1) DWORDs before each pad insertion
//   pad_amt_code: pad = (code+1) DWORDs
// D# layout per CDNA5 ISA sec. 8.3/8.4; groups 2,3 zero (2D tensor), cpol 0.
// This toolchain's builtin is the 6-arg form:
//   (u32x4 g0, i32x8 g1, i32x4 g2, i32x4 g3, i32x8 extra, i32 cpol)
// Issue from one wave only; completion via s_wait_tensorcnt.
// ---------------------------------------------------------------------------
__device__ __forceinline__ unsigned lds_off(const void* p) {
  // generic pointer to LDS: low 32 bits are the LDS byte offset
  return (unsigned)(uintptr_t)p;
}

__device__ __forceinline__ void tdm_load_2d(
    unsigned dst_lds, const void* src_global,
    unsigned tile_w, unsigned tile_h, unsigned long long stride_elems,
    unsigned pad_int_code, unsigned pad_amt_code)
{
  unsigned long long ga = (unsigned long long)(uintptr_t)src_global;
  u32x4 g0;
  g0[0] = 1u;                                          // count=1, user mode
  g0[1] = dst_lds;                                     // lds_addr
  g0[2] = (unsigned)ga;                                // global_addr[31:0]
  g0[3] = (unsigned)((ga >> 32) & 0x01FFFFFFu)         // global_addr[56:32]
        | (2u << 30);                                  // type = 2 ("image")
  i32x8v g1;
  g1[0] = (int)((2u << 16)                             // data_size = 4 bytes
              | (1u << 20)                             // pad_enable
              | (pad_int_code << 22)
              | (pad_amt_code << 25));
  g1[1] = (int)((tile_w & 0xFFFFu) << 16);             // tensor_dim0 lo16 (dim0 = tile_w)
  g1[2] = (int)(((tile_w >> 16) & 0xFFFFu)             // tensor_dim0 hi16
              | ((tile_h & 0xFFFFu) << 16));           // tensor_dim1 lo16 (dim1 = tile_h)
  g1[3] = (int)(((tile_h >> 16) & 0xFFFFu)             // tensor_dim1 hi16
              | (tile_w << 16));                       // tile_dim0
  g1[4] = (int)(tile_h & 0xFFFFu);                     // tile_dim1 (tile_dim2 = 0)
  g1[5] = (int)(stride_elems & 0xFFFFFFFFull);         // tensor_dim0_stride lo32
  g1[6] = (int)((stride_elems >> 32) & 0xFFFFull);     // stride hi16 (dim1_stride = 0)
  g1[7] = 0;
  i32x4v gz4 = (i32x4v)0;
  i32x8v gz8 = (i32x8v)0;
  __builtin_amdgcn_tensor_load_to_lds(g0, g1, gz4, gz4, gz8, 0);
}

// ---------------------------------------------------------------------------
// Kernel A: per-n masked rank-B update  so[n][d][e] = sum_b m*E_d*E_e (partials)
// grid (Ncl, SLICES), block 256 (8 waves). Wave w owns d-tile row d0=16*w.
// Embedding chunks staged to LDS via TDM, double-buffered.
// ---------------------------------------------------------------------------
__global__ __launch_bounds__(256) void k_accum(
    const float* __restrict__ emb, const long long* __restrict__ states,
    float* __restrict__ so_part, float* __restrict__ sv_part,
    float* __restrict__ cnt_part)
{
  __shared__ float sE[2][CHUNKA * EPITCH];
  __shared__ float sM[2][CHUNKA];
  __shared__ float sRed[CHUNKA];

  const int tid   = threadIdx.x;
  const int n     = blockIdx.x;
  const int slice = blockIdx.y;
  const int lane  = tid & 31;
  const int wave  = tid >> 5;
  const int lmod  = lane & 15;
  const int lhalf = lane >> 4;

  v8f acc[8];
#pragma unroll
  for (int i = 0; i < 8; ++i) acc[i] = (v8f)0.0f;

  float sv_acc  = 0.0f;
  float cnt_acc = 0.0f;

  const int b_begin = slice * (Bsz / SLICES);
  const int nchunks = (Bsz / SLICES) / CHUNKA;

  // prologue: stage chunk 0
  if (wave == 0) {
    tdm_load_2d(lds_off(&sE[0][0]),
                emb + ((size_t)b_begin * Ncl + n) * Dd,
                Dd, CHUNKA, (unsigned long long)Ncl * Dd, 6u, 7u);
  }
  if (tid < CHUNKA) {
    float m = (states[(size_t)(b_begin + tid) * Ncl + n] > 0) ? 1.0f : 0.0f;
    sM[0][tid] = m;
    cnt_acc += m;
  }
  if (wave == 0) __builtin_amdgcn_s_wait_tensorcnt(0);
  __syncthreads();

  for (int ch = 0; ch < nchunks; ++ch) {
    const int cur = ch & 1;

    // prefetch next chunk (TDM into other buffer) while computing this one
    if (ch + 1 < nchunks) {
      const int bn = b_begin + (ch + 1) * CHUNKA;
      if (wave == 0) {
        tdm_load_2d(lds_off(&sE[cur ^ 1][0]),
                    emb + ((size_t)bn * Ncl + n) * Dd,
                    Dd, CHUNKA, (unsigned long long)Ncl * Dd, 6u, 7u);
      }
      if (tid < CHUNKA) {
        float m = (states[(size_t)(bn + tid) * Ncl + n] > 0) ? 1.0f : 0.0f;
        sM[cur ^ 1][tid] = m;
        cnt_acc += m;
      }
    }

    const float* E = &sE[cur][0];
    const float* M = &sM[cur][0];

    // masked vector sum (threads 0..127 own one column each)
    if (tid < Dd) {
      float a = 0.0f;
#pragma unroll 4
      for (int i = 0; i < CHUNKA; ++i) a = fmaf(M[i], E[i * EPITCH + tid], a);
      sv_acc += a;
    }

    // WMMA: C[d,e] += sum_k mE[k,d] * E[k,e], K in steps of 4
    const int d0 = wave * 16;
#pragma unroll 2
    for (int kb = 0; kb < CHUNKA / 4; ++kb) {
      const int r0 = 4 * kb + 2 * lhalf;      // this lane's two K-rows
      const float m0 = M[r0];
      const float m1 = M[r0 + 1];
      v2f a;
      a.x = m0 * E[r0 * EPITCH + d0 + lmod];
      a.y = m1 * E[(r0 + 1) * EPITCH + d0 + lmod];
#pragma unroll
      for (int et = 0; et < 8; ++et) {
        v2f b;
        b.x = E[r0 * EPITCH + 16 * et + lmod];
        b.y = E[(r0 + 1) * EPITCH + 16 * et + lmod];
        acc[et] = __builtin_amdgcn_wmma_f32_16x16x4_f32(
            false, a, false, b, (short)0, acc[et], false, false);
      }
    }

    if (wave == 0) __builtin_amdgcn_s_wait_tensorcnt(0);
    __syncthreads();
  }

  // store 128x128 partial: slot (v,l) -> d = 16w+v+8*lhalf, e = 16et+lmod
#pragma unroll
  for (int et = 0; et < 8; ++et) {
#pragma unroll
    for (int v = 0; v < 8; ++v) {
      int d = wave * 16 + v + 8 * lhalf;
      int e = 16 * et + lmod;
      so_part[(((size_t)slice * Ncl + n) * Dd + d) * Dd + e] = acc[et][v];
    }
  }
  if (tid < Dd) sv_part[((size_t)slice * Ncl + n) * Dd + tid] = sv_acc;
  if (tid < CHUNKA) sRed[tid] = cnt_acc;
  __syncthreads();
  if (tid == 0) {
    float c = 0.0f;
    for (int i = 0; i < CHUNKA; ++i) c += sRed[i];
    cnt_part[slice * Ncl + n] = c;
  }
}

// ---------------------------------------------------------------------------
// Kernel B: reduce partials, build cov, Gauss-Jordan inverse in 131.5KB LDS.
// grid Ncl blocks, 256 threads, dynamic LDS = 128*257*4 bytes.
// ---------------------------------------------------------------------------
__global__ __launch_bounds__(256) void k_invert(
    const float* __restrict__ so_in, const long long* __restrict__ count_in,
    const float* __restrict__ sv_in,
    const float* __restrict__ so_part, const float* __restrict__ sv_part,
    const float* __restrict__ cnt_part,
    float* __restrict__ mean_out, float* __restrict__ cnt_out,
    float* __restrict__ inv_out)
{
  extern __shared__ float aug[];          // [Dd][AUGP]
  __shared__ float sMean[Dd];
  __shared__ float sC[1];

  const int tid = threadIdx.x;
  const int n   = blockIdx.x;

  if (tid == 0) {
    float c = (float)count_in[n];
    for (int s = 0; s < SLICES; ++s) c += cnt_part[s * Ncl + n];
    cnt_out[n] = c;
    sC[0] = fmaxf(c, 1.0f);
  }
  __syncthreads();
  const float safe_c = sC[0];
  const float inv_c  = 1.0f / safe_c;

  if (tid < Dd) {
    float sv = sv_in[(size_t)n * Dd + tid];
    for (int s = 0; s < SLICES; ++s) sv += sv_part[((size_t)s * Ncl + n) * Dd + tid];
    float m = sv * inv_c;
    sMean[tid] = m;
    mean_out[(size_t)n * Dd + tid] = m;
  }
  __syncthreads();

  // build augmented [cov | I]
  for (int idx = tid; idx < Dd * 256; idx += 256) {
    int r = idx >> 8;
    int c = idx & 255;
    float val;
    if (c < Dd) {
      float so = so_in[((size_t)n * Dd + r) * Dd + c];
      for (int s = 0; s < SLICES; ++s)
        so += so_part[(((size_t)s * Ncl + n) * Dd + r) * Dd + c];
      val = so * inv_c - sMean[r] * sMean[c] + ((r == c) ? EPSV : 0.0f);
    } else {
      val = ((c - Dd) == r) ? 1.0f : 0.0f;
    }
    aug[r * AUGP + c] = val;
  }
  __syncthreads();

  // pivot-free Gauss-Jordan (SPD); active columns at step k are k .. k+128
  const int i = tid & 127;
  const int h = tid >> 7;
  for (int k = 0; k < Dd; ++k) {
    float ip = 1.0f / aug[k * AUGP + k];
    __syncthreads();
    if (tid <= Dd) aug[k * AUGP + k + tid] *= ip;   // 129 active cols
    __syncthreads();
    float f = aug[i * AUGP + k];
    __syncthreads();
    if (i != k) {
      int cbase = k + h * 65;
      int cnum  = h ? 64 : 65;
      for (int j = 0; j < cnum; ++j) {
        int c = cbase + j;
        aug[i * AUGP + c] = fmaf(-f, aug[k * AUGP + c], aug[i * AUGP + c]);
      }
    }
    __syncthreads();
  }

  for (int idx = tid; idx < Dd * Dd; idx += 256) {
    int r = idx >> 7;
    int c = idx & 127;
    inv_out[((size_t)n * Dd + r) * Dd + c] = aug[r * AUGP + Dd + c];
  }
}

// ---------------------------------------------------------------------------
// Kernel C: d2[b,n] = diff . inv . diff via WMMA, fused row-dot epilogue.
// grid (Ncl, Bsz/128), block 256 (8 waves x 16 b-rows).
// inv and raw-embedding chunks staged via TDM; mean subtracted at use.
// ---------------------------------------------------------------------------
__global__ __launch_bounds__(256) void k_mahal(
    const float* __restrict__ emb, const float* __restrict__ inv,
    const float* __restrict__ mean, const float* __restrict__ cntv,
    float* __restrict__ out)
{
  __shared__ float sInv[32 * IPITCH];
  __shared__ float sEmb[Dd * DPITCH];   // raw embedding chunk (mean applied at use)
  __shared__ float sMean[Dd];

  const int tid   = threadIdx.x;
  const int n     = blockIdx.x;
  const int bbase = blockIdx.y * 128;
  const int lane  = tid & 31;
  const int wave  = tid >> 5;
  const int lmod  = lane & 15;
  const int lhalf = lane >> 4;

  if (tid < Dd) sMean[tid] = mean[(size_t)n * Dd + tid];
  const float c = cntv[n];
  const bool ready = (c >= MINC);

  v8f acc[8];
#pragma unroll
  for (int i = 0; i < 8; ++i) acc[i] = (v8f)0.0f;

  // T[b,e] = sum_d diff[b,d] * inv[d,e], D chunked by 32
  for (int dc = 0; dc < 4; ++dc) {
    const int d0 = dc * 32;
    if (wave == 0) {
      // inv rows d0..d0+31 (contiguous 32x128), pad -> pitch 136
      tdm_load_2d(lds_off(sInv), inv + ((size_t)n * Dd + d0) * Dd,
                  Dd, 32, (unsigned long long)Dd, 6u, 7u);
      // emb rows bbase..bbase+127, cols d0..d0+31 (strided), pad -> pitch 36
      tdm_load_2d(lds_off(sEmb), emb + (((size_t)bbase * Ncl + n) * Dd + d0),
                  32, Dd, (unsigned long long)Ncl * Dd, 4u, 3u);
      __builtin_amdgcn_s_wait_tensorcnt(0);
    }
    __syncthreads();

#pragma unroll 2
    for (int k = 0; k < 8; ++k) {
      const int dl = 4 * k + 2 * lhalf;
      v2f raw = *(const v2f*)(sEmb + (16 * wave + lmod) * DPITCH + dl);
      v2f ms;
      ms.x = sMean[d0 + dl];
      ms.y = sMean[d0 + dl + 1];
      v2f a = raw - ms;
#pragma unroll
      for (int et = 0; et < 8; ++et) {
        v2f b;
        b.x = sInv[dl * IPITCH + 16 * et + lmod];
        b.y = sInv[(dl + 1) * IPITCH + 16 * et + lmod];
        acc[et] = __builtin_amdgcn_wmma_f32_16x16x4_f32(
            false, a, false, b, (short)0, acc[et], false, false);
      }
    }
    __syncthreads();
  }

  // epilogue: d2 partial = sum_e T[b,e] * diff[b,e]
  float part[8];
#pragma unroll
  for (int v = 0; v < 8; ++v) part[v] = 0.0f;

  for (int ec = 0; ec < 4; ++ec) {
    const int e0 = ec * 32;
    if (wave == 0) {
      tdm_load_2d(lds_off(sEmb), emb + (((size_t)bbase * Ncl + n) * Dd + e0),
                  32, Dd, (unsigned long long)Ncl * Dd, 4u, 3u);
      __builtin_amdgcn_s_wait_tensorcnt(0);
    }
    __syncthreads();
#pragma unroll
    for (int tt = 0; tt < 2; ++tt) {
      const int et = 2 * ec + tt;
      const int el = 16 * tt + lmod;
#pragma unroll
      for (int v = 0; v < 8; ++v) {
        int row = 16 * wave + v + 8 * lhalf;
        float d = sEmb[row * DPITCH + el] - sMean[e0 + el];
        part[v] = fmaf(acc[et][v], d, part[v]);
      }
    }
    __syncthreads();
  }

  // reduce across the 16 lanes of each N-group (wave32), write output
#pragma unroll
  for (int v = 0; v < 8; ++v) {
    float x = part[v];
    x += __shfl_xor(x, 1, 32);
    x += __shfl_xor(x, 2, 32);
    x += __shfl_xor(x, 4, 32);
    x += __shfl_xor(x, 8, 32);
    if (lmod == 0) {
      int b = bbase + 16 * wave + v + 8 * lhalf;
      out[(size_t)b * Ncl + n] = ready ? sqrtf(fmaxf(x, 0.0f)) : 0.0f;
    }
  }
}

// ---------------------------------------------------------------------------
extern "C" void kernel_launch(void* const* d_in, const int* in_sizes, int n_in,
                              void* d_out, int out_size, void* d_ws, size_t ws_size,
                              hipStream_t stream)
{
  const float*     emb       = (const float*)d_in[0];
  const long long* states    = (const long long*)d_in[1];
  const float*     sum_vec   = (const float*)d_in[2];
  const float*     sum_outer = (const float*)d_in[3];
  const long long* count     = (const long long*)d_in[4];
  float*           out       = (float*)d_out;

  float* ws      = (float*)d_ws;
  float* ws_so   = ws;                                     // SLICES*64*128*128
  float* ws_sv   = ws_so + (size_t)SLICES * Ncl * Dd * Dd; // SLICES*64*128
  float* ws_cnt  = ws_sv + (size_t)SLICES * Ncl * Dd;      // SLICES*64
  float* ws_mean = ws_cnt + SLICES * Ncl;                  // 64*128
  float* ws_cntT = ws_mean + (size_t)Ncl * Dd;             // 64
  float* ws_inv  = ws_cntT + Ncl;                          // 64*128*128

  k_accum<<<dim3(Ncl, SLICES), 256, 0, stream>>>(emb, states, ws_so, ws_sv, ws_cnt);
  k_invert<<<dim3(Ncl), 256, Dd * AUGP * sizeof(float), stream>>>(
      sum_outer, count, sum_vec, ws_so, ws_sv, ws_cnt, ws_mean, ws_cntT, ws_inv);
  k_mahal<<<dim3(Ncl, Bsz / 128), 256, 0, stream>>>(emb, ws_inv, ws_mean, ws_cntT, out);
}